// MixtureOfExperts_38809324487362
// MI455X (gfx1250) — compile-verified
//
#include <hip/hip_runtime.h>
#include <hip/hip_bf16.h>

typedef float v2f __attribute__((ext_vector_type(2)));
typedef float v8f __attribute__((ext_vector_type(8)));

#define B_DIM 8192
#define P_DIM 1024
#define Q_DIM 1024
#define E_DIM 8

#define BM 128
#define BN 128
#define BK 32
#define NKT ((E_DIM * P_DIM) / BK)      // 256 K-tiles

#define ASTRIDE 36                       // padded A row stride (floats): conflict-free frags
#define BSTRIDE 144                      // padded B row stride (floats): conflict-free frags
#define ABUF (BM * ASTRIDE)              // 4608 floats per A buffer
#define BBUF (BK * BSTRIDE)              // 4608 floats per B buffer
#define SMEM_FLOATS (2 * ABUF + 2 * BBUF + BM * E_DIM)
#define SMEM_BYTES (SMEM_FLOATS * 4)     // 77824 B (dynamic LDS; WGP has 320 KB)

// ---- CDNA5 async global->LDS copy (no VGPR staging, tracked by ASYNCcnt) ----
static __device__ __forceinline__ void async_cp16(float* lds, const float* g) {
#if __has_builtin(__builtin_amdgcn_global_load_async_to_lds_b128)
    typedef int vs4i __attribute__((vector_size(16)));   // matches builtin's V4i
    const vs4i* gp = (const vs4i*)g;                     // generic AS, reinterpret
    vs4i*       lp = (vs4i*)lds;
    __builtin_amdgcn_global_load_async_to_lds_b128(
        (__attribute__((address_space(1))) vs4i*)gp,     // global src
        (__attribute__((address_space(3))) vs4i*)lp,     // LDS dst
        0, 0);
#else
    *(float4*)lds = *(const float4*)g;   // fallback: synchronous copy
#endif
}

static __device__ __forceinline__ void wait_async() {
#if __has_builtin(__builtin_amdgcn_global_load_async_to_lds_b128)
#if __has_builtin(__builtin_amdgcn_s_wait_asynccnt)
    __builtin_amdgcn_s_wait_asynccnt(0);
#else
    asm volatile("s_wait_asynccnt 0x0" ::: "memory");
#endif
#endif
}

// ---------------------------------------------------------------------------
// Kernel 1: router  weights[b,e] = softmax(x[b,:] @ Wg + bg); one wave / row
// ---------------------------------------------------------------------------
__global__ __launch_bounds__(256) void moe_router(
    const float* __restrict__ inp, const float* __restrict__ Wg,
    const float* __restrict__ bg, float* __restrict__ wts)
{
    const int lane = threadIdx.x & 31;
    const int wid  = threadIdx.x >> 5;
    const int row  = blockIdx.x * 8 + wid;

    const float* __restrict__ x = inp + (size_t)row * P_DIM;

    float acc[E_DIM];
#pragma unroll
    for (int e = 0; e < E_DIM; ++e) acc[e] = 0.f;

    for (int p = lane; p < P_DIM; p += 32) {
        float xv = x[p];
        const float4* g = (const float4*)(Wg + (size_t)p * E_DIM);
        float4 g0 = g[0], g1 = g[1];
        acc[0] += xv * g0.x; acc[1] += xv * g0.y;
        acc[2] += xv * g0.z; acc[3] += xv * g0.w;
        acc[4] += xv * g1.x; acc[5] += xv * g1.y;
        acc[6] += xv * g1.z; acc[7] += xv * g1.w;
    }

#pragma unroll
    for (int e = 0; e < E_DIM; ++e) {
#pragma unroll
        for (int off = 16; off > 0; off >>= 1)
            acc[e] += __shfl_xor(acc[e], off, 32);
    }

    float mx = -3.402823466e+38f;
#pragma unroll
    for (int e = 0; e < E_DIM; ++e) { acc[e] += bg[e]; mx = fmaxf(mx, acc[e]); }
    float s = 0.f;
#pragma unroll
    for (int e = 0; e < E_DIM; ++e) { acc[e] = __expf(acc[e] - mx); s += acc[e]; }
    float inv = 1.f / s;

    if (lane == 0) {
        float4 w0 = make_float4(acc[0]*inv, acc[1]*inv, acc[2]*inv, acc[3]*inv);
        float4 w1 = make_float4(acc[4]*inv, acc[5]*inv, acc[6]*inv, acc[7]*inv);
        float4* wp = (float4*)(wts + (size_t)row * E_DIM);
        wp[0] = w0; wp[1] = w1;
    }
}

// ---------------------------------------------------------------------------
// Kernel 2: aux loss = E * sum_e (mean_b w[b,e])^2   (deterministic)
// ---------------------------------------------------------------------------
__global__ __launch_bounds__(256) void moe_aux(
    const float* __restrict__ wts, float* __restrict__ out_aux)
{
    __shared__ float red[256][E_DIM];
    const int tid = threadIdx.x;

    float loc[E_DIM];
#pragma unroll
    for (int e = 0; e < E_DIM; ++e) loc[e] = 0.f;

    for (int r = tid; r < B_DIM; r += 256) {
        const float4* wrow = (const float4*)(wts + (size_t)r * E_DIM);
        float4 a = wrow[0], b = wrow[1];
        loc[0] += a.x; loc[1] += a.y; loc[2] += a.z; loc[3] += a.w;
        loc[4] += b.x; loc[5] += b.y; loc[6] += b.z; loc[7] += b.w;
    }
#pragma unroll
    for (int e = 0; e < E_DIM; ++e) red[tid][e] = loc[e];
    __syncthreads();

    for (int s = 128; s > 0; s >>= 1) {
        if (tid < s) {
#pragma unroll
            for (int e = 0; e < E_DIM; ++e) red[tid][e] += red[tid + s][e];
        }
        __syncthreads();
    }

    if (tid == 0) {
        float aux = 0.f;
#pragma unroll
        for (int e = 0; e < E_DIM; ++e) {
            float m = red[0][e] * (1.f / (float)B_DIM);
            aux += m * m;
        }
        out_aux[0] = (float)E_DIM * aux;
    }
}

// ---------------------------------------------------------------------------
// Kernel 3: fused MoE GEMM, K = E*P = 8192.
//   out[b,q] = sum_{e,p} (w[b,e]*x[b,p]) * W[e,p,q] + sum_e w[b,e]*bias[e,q]
// Router weight applied to A *fragments* (per-lane row scalar), so tiles can
// stream straight to LDS via GLOBAL_LOAD_ASYNC_TO_LDS_B128 (zero staging
// VGPRs, ASYNCcnt-tracked double buffering). 128x128 block tile, 8 waves
// (4M x 2N), each wave = 32x64 patch = 8 V_WMMA_F32_16X16X4_F32 accumulators.
// ---------------------------------------------------------------------------
__global__ __launch_bounds__(256) void moe_gemm(
    const float* __restrict__ inp,   // [B,P]
    const float* __restrict__ W,     // [E,P,Q]
    const float* __restrict__ bias,  // [E,Q]
    const float* __restrict__ wts,   // [B,E]
    float* __restrict__ out)         // [B,Q]
{
    extern __shared__ float smem[];
    float* As = smem;                        // 2 x [BM][ASTRIDE]
    float* Bs = smem + 2 * ABUF;             // 2 x [BK][BSTRIDE]
    float* Ws = smem + 2 * ABUF + 2 * BBUF;  // [BM][E_DIM]

    const int tid   = threadIdx.x;
    const int lane  = tid & 31;
    const int wid   = tid >> 5;
    const int warpM = wid & 3;               // rows warpM*32
    const int warpN = wid >> 2;              // cols warpN*64
    const int rowBase = blockIdx.y * BM;
    const int colBase = blockIdx.x * BN;

    // stage router weights for this block's 128 rows (1 float4 per thread)
    ((float4*)Ws)[tid] = ((const float4*)(wts + (size_t)rowBase * E_DIM))[tid];

    v8f acc[2][4] = {};                      // [mi][ni]

    const int mrow0 = warpM * 32 + (lane & 15);
    const int klo   = (lane >> 4) << 1;      // lanes 0-15: K{0,1}; 16-31: K{2,3}
    const int ncol  = warpN * 64 + (lane & 15);
    const int hi    = lane >> 4;

    // issue async copies of tile kt into LDS buffer `buf`
    auto issueTile = [&](int kt, int buf) {
        const int kglob = kt * BK;
        const int e  = kglob >> 10;
        const int pb = kglob & (P_DIM - 1);
        float* Ab = As + buf * ABUF;
        float* Bb = Bs + buf * BBUF;
#pragma unroll
        for (int i = 0; i < 4; ++i) {
            int idx = tid + i * 256;                 // 0..1023
            int m   = idx >> 3;                      // 8 float4 per A row
            int k4  = (idx & 7) << 2;
            async_cp16(Ab + m * ASTRIDE + k4,
                       inp + (size_t)(rowBase + m) * P_DIM + pb + k4);
        }
#pragma unroll
        for (int i = 0; i < 4; ++i) {
            int idx = tid + i * 256;                 // 0..1023
            int k   = idx >> 5;                      // 32 float4 per B row
            int n4  = (idx & 31) << 2;
            async_cp16(Bb + k * BSTRIDE + n4,
                       W + (size_t)e * P_DIM * Q_DIM
                         + (size_t)(pb + k) * Q_DIM + colBase + n4);
        }
    };

    issueTile(0, 0);
    wait_async();
    __syncthreads();                         // tile 0 + Ws visible

#pragma unroll 2
    for (int kt = 0; kt < NKT; ++kt) {
        const int cur = kt & 1;
        if (kt + 1 < NKT) issueTile(kt + 1, cur ^ 1);   // overlap with WMMA

        const int e = kt >> 5;               // expert of this tile (BK=32)
        const float w0 = Ws[mrow0 * E_DIM + e];
        const float w1 = Ws[(mrow0 + 16) * E_DIM + e];

        const float* A0 = As + cur * ABUF;
        const float* B0 = Bs + cur * BBUF;
#pragma unroll
        for (int kk = 0; kk < BK / 4; ++kk) {
            const int k0 = kk * 4 + klo;
            v2f a0 = *(const v2f*)(A0 + mrow0 * ASTRIDE + k0);
            v2f a1 = *(const v2f*)(A0 + (mrow0 + 16) * ASTRIDE + k0);
            a0.x *= w0; a0.y *= w0;          // fold router weight into A frag
            a1.x *= w1; a1.y *= w1;
            v2f b[4];
#pragma unroll
            for (int ni = 0; ni < 4; ++ni) {
                b[ni].x = B0[k0 * BSTRIDE + ncol + ni * 16];
                b[ni].y = B0[(k0 + 1) * BSTRIDE + ncol + ni * 16];
            }
#pragma unroll
            for (int ni = 0; ni < 4; ++ni) {
                acc[0][ni] = __builtin_amdgcn_wmma_f32_16x16x4_f32(
                    false, a0, false, b[ni], (short)0, acc[0][ni], false, false);
                acc[1][ni] = __builtin_amdgcn_wmma_f32_16x16x4_f32(
                    false, a1, false, b[ni], (short)0, acc[1][ni], false, false);
            }
        }

        if (kt + 1 < NKT) {
            wait_async();                    // next tile fully landed in LDS
            __syncthreads();
        }
    }

    // epilogue: out = acc + sum_e w[row,e]*bias[e,q]
#pragma unroll
    for (int ni = 0; ni < 4; ++ni) {
        const int q = colBase + warpN * 64 + ni * 16 + (lane & 15);
        float bv[E_DIM];
#pragma unroll
        for (int e2 = 0; e2 < E_DIM; ++e2)
            bv[e2] = bias[(size_t)e2 * Q_DIM + q];

#pragma unroll
        for (int mi = 0; mi < 2; ++mi) {
#pragma unroll
            for (int i = 0; i < 8; ++i) {
                const int rloc = warpM * 32 + mi * 16 + hi * 8 + i;
                float extra = 0.f;
#pragma unroll
                for (int e2 = 0; e2 < E_DIM; ++e2)
                    extra += Ws[rloc * E_DIM + e2] * bv[e2];
                out[(size_t)(rowBase + rloc) * Q_DIM + q] = acc[mi][ni][i] + extra;
            }
        }
    }
}

// ---------------------------------------------------------------------------
extern "C" void kernel_launch(void* const* d_in, const int* in_sizes, int n_in,
                              void* d_out, int out_size, void* d_ws, size_t ws_size,
                              hipStream_t stream) {
    const float* inp  = (const float*)d_in[0];   // [8192,1024]
    const float* W    = (const float*)d_in[1];   // [8,1024,1024]
    const float* bias = (const float*)d_in[2];   // [8,1024]
    const float* Wg   = (const float*)d_in[3];   // [1024,8]
    const float* bg   = (const float*)d_in[4];   // [8]

    float* out = (float*)d_out;                  // [8192*1024] output + [1] aux
    float* wts = (float*)d_ws;                   // weights [8192,8] scratch

    moe_router<<<B_DIM / 8, 256, 0, stream>>>(inp, Wg, bg, wts);
    moe_aux<<<1, 256, 0, stream>>>(wts, out + (size_t)B_DIM * Q_DIM);

    dim3 grid(Q_DIM / BN, B_DIM / BM);           // (8, 64)
    moe_gemm<<<grid, 256, SMEM_BYTES, stream>>>(inp, W, bias, wts, out);
}